// L2_76338748719714
// MI455X (gfx1250) — compile-verified
//
#include <hip/hip_runtime.h>
#include <stdint.h>

// ---------------------------------------------------------------------------
// Per-row L2 normalization, (ROWS x 4096) fp32.
// Memory-bound streaming kernel: target = HBM roofline (~23 us @ 23.3 TB/s).
// CDNA5 path: async global->LDS DMA (ASYNCcnt), wave32 shuffle reduction,
// 128-bit accesses, nontemporal stores.
// ---------------------------------------------------------------------------

#define ROW_LEN   4096                 // floats per row
#define BLOCK     256                  // threads per block = 8 waves (wave32)
#define VEC       4                    // floats per b128 access
#define CHUNK     (BLOCK * VEC)        // 1024 floats covered per pass
#define NPASS     (ROW_LEN / CHUNK)    // 4 passes to cover one row
#define NWAVES    (BLOCK / 32)         // 8

typedef __attribute__((ext_vector_type(4))) float v4f;   // native vector (NT-store capable)
typedef __attribute__((ext_vector_type(4))) int   v4i;   // builtin's element type

// Pointer-to-addrspace-qualified-v4i types matching the builtin prototype:
//   void __builtin_amdgcn_global_load_async_to_lds_b128(v4i as(1)*, v4i as(3)*, Ii, Ii)
typedef __attribute__((address_space(1))) v4i* gv4i_ptr;
typedef __attribute__((address_space(3))) v4i* lv4i_ptr;

#if __has_builtin(__builtin_amdgcn_global_load_async_to_lds_b128)
  #define HAVE_ASYNC_BUILTIN 1
#else
  #define HAVE_ASYNC_BUILTIN 0
  #warning "no __builtin_amdgcn_global_load_async_to_lds_b128 -- probing inline asm form"
#endif

__global__ __launch_bounds__(BLOCK) void l2norm_rows_kernel(
    const float* __restrict__ in, float* __restrict__ out)
{
  __shared__ __align__(16) float lrow[ROW_LEN];   // 16 KB row staging
  __shared__ float wsum[NWAVES];

  const int    t       = threadIdx.x;
  const size_t rowbase = (size_t)blockIdx.x * (size_t)ROW_LEN;
  const float* __restrict__ rin  = in  + rowbase;
  float*       __restrict__ rout = out + rowbase;

  // ---- Stage 1: async DMA the whole row global -> LDS (CDNA5 ASYNCcnt path)
  #pragma unroll
  for (int i = 0; i < NPASS; ++i) {
    const int off = i * CHUNK + t * VEC;          // float offset, 16B aligned
#if HAVE_ASYNC_BUILTIN
    __builtin_amdgcn_global_load_async_to_lds_b128(
        (gv4i_ptr)(rin + off), (lv4i_ptr)(lrow + off), /*offset=*/0, /*cpol=*/0);
#else
    uint32_t lds_addr = (uint32_t)(uintptr_t)(lrow + off);
    const float* gaddr = rin + off;
    asm volatile("global_load_async_to_lds_b128 %0, %1, off"
                 :: "v"(lds_addr), "v"(gaddr) : "memory");
#endif
  }
#if __has_builtin(__builtin_amdgcn_s_wait_asynccnt)
  __builtin_amdgcn_s_wait_asynccnt(0);
#else
  asm volatile("s_wait_asynccnt 0" ::: "memory");
#endif
  __syncthreads();   // make all waves' LDS writes visible to everyone

  // ---- Stage 2: sum of squares (ds_load_b128 from LDS, keep data in regs)
  v4f   v[NPASS];
  float acc = 0.0f;
  #pragma unroll
  for (int i = 0; i < NPASS; ++i) {
    const int off = i * CHUNK + t * VEC;
    v[i] = *reinterpret_cast<const v4f*>(lrow + off);
    acc += v[i].x * v[i].x;
    acc += v[i].y * v[i].y;
    acc += v[i].z * v[i].z;
    acc += v[i].w * v[i].w;
  }

  // wave32 butterfly reduction (5 steps)
  #pragma unroll
  for (int m = 16; m > 0; m >>= 1)
    acc += __shfl_xor(acc, m, 32);

  const int wave = t >> 5;
  const int lane = t & 31;
  if (lane == 0) wsum[wave] = acc;
  __syncthreads();

  float total = 0.0f;
  #pragma unroll
  for (int w = 0; w < NWAVES; ++w) total += wsum[w];

  const float scale = 1.0f / sqrtf(total);

  // ---- Stage 3: scale + nontemporal b128 stores (output never re-read)
  #pragma unroll
  for (int i = 0; i < NPASS; ++i) {
    const int off = i * CHUNK + t * VEC;
    v4f o;
    o.x = v[i].x * scale;
    o.y = v[i].y * scale;
    o.z = v[i].z * scale;
    o.w = v[i].w * scale;
    __builtin_nontemporal_store(o, reinterpret_cast<v4f*>(rout + off));
  }
}

extern "C" void kernel_launch(void* const* d_in, const int* in_sizes, int n_in,
                              void* d_out, int out_size, void* d_ws, size_t ws_size,
                              hipStream_t stream) {
  (void)n_in; (void)out_size; (void)d_ws; (void)ws_size;
  const float* in  = (const float*)d_in[0];
  float*       out = (float*)d_out;
  const int rows = in_sizes[0] / ROW_LEN;   // 16384 for the reference shapes
  l2norm_rows_kernel<<<dim3(rows), dim3(BLOCK), 0, stream>>>(in, out);
}